// TripletLoss_80676665688070
// MI455X (gfx1250) — compile-verified
//
#include <hip/hip_runtime.h>

typedef float v2f __attribute__((ext_vector_type(2)));
typedef float v8f __attribute__((ext_vector_type(8)));

#define NN 8192
#define DD 128
#define MARGIN_F 0.3f
#define ROWS_PER_BLOCK 128
#define COL_SPLIT 8
#define COLS_PER_BLOCK (NN / COL_SPLIT)   /* 1024 */
#define COL_TILES (COLS_PER_BLOCK / 16)   /* 64   */
#define LDS_STRIDE 132                    /* 128 + 4 pad -> conflict-free */

// ---- order-preserving float <-> uint encoding (for integer atomics) ----
__device__ __forceinline__ unsigned ford(float f) {
  unsigned u = __float_as_uint(f);
  return (u & 0x80000000u) ? ~u : (u | 0x80000000u);
}
__device__ __forceinline__ float forddec(unsigned u) {
  unsigned b = (u & 0x80000000u) ? (u ^ 0x80000000u) : ~u;
  return __uint_as_float(b);
}

__global__ void th_init(unsigned* posU, unsigned* negU, float* out) {
  int i = blockIdx.x * 256 + threadIdx.x;
  if (i == 0) out[0] = 0.0f;
  if (i < NN) {
    posU[i] = ford(-3.0e38f);
    negU[i] = ford(3.0e38f);
  }
}

__global__ __launch_bounds__(256)
void th_main(const float* __restrict__ E, const int* __restrict__ labels,
             unsigned* __restrict__ posU, unsigned* __restrict__ negU) {
  __shared__ float bt[2][16 * LDS_STRIDE];   // double-buffered B tile
  __shared__ int   lbl[COLS_PER_BLOCK];      // this block's column labels

  const int lane = threadIdx.x & 31;
  const int wave = threadIdx.x >> 5;
  const int lh   = lane >> 4;       // lane-half: 0 or 1
  const int lcol = lane & 15;       // N index within tile / M index for A
  const int m0   = blockIdx.x * ROWS_PER_BLOCK + wave * 16;
  const int cbase = blockIdx.y * COLS_PER_BLOCK;
  const int koff = lh * 2;          // A/B layout: K = 4*s + 2*half + {0,1}

  // ---- stage this block's 1024 column labels into LDS (once) ----
  {
    int4 l4 = *(const int4*)(labels + cbase + threadIdx.x * 4);
    *(int4*)(&lbl[threadIdx.x * 4]) = l4;
  }

  // ---- A tile (16 x 128) resident in registers, exact WMMA A layout ----
  const float* arow = E + (size_t)(m0 + lcol) * DD + koff;
  v2f a[32];
#pragma unroll
  for (int s = 0; s < 32; ++s) {
    a[s].x = arow[4 * s];
    a[s].y = arow[4 * s + 1];
  }

  int li[8];
#pragma unroll
  for (int v = 0; v < 8; ++v) li[v] = labels[m0 + 8 * lh + v];

  float posr[8], negr[8];
#pragma unroll
  for (int v = 0; v < 8; ++v) { posr[v] = -3.0e38f; negr[v] = 3.0e38f; }

  const int brow_ld = threadIdx.x >> 4;       // 0..15 (tile row to stage)
  const int bk_ld   = (threadIdx.x & 15) * 8; // 0..120

  // ---- software pipeline prologue: fetch tile 0 into registers ----
  const float* src0 = E + (size_t)(cbase + brow_ld) * DD + bk_ld;
  float4 p0 = *(const float4*)(src0);
  float4 p1 = *(const float4*)(src0 + 4);

  for (int ct = 0; ct < COL_TILES; ++ct) {
    const int buf = ct & 1;

    // stage current tile's registers into this iteration's LDS buffer
    // (the implicit s_wait_loadcnt here is the ONLY drain point for the
    //  B-tile loads issued one iteration ago, after a full WMMA chain)
    {
      float* dst = &bt[buf][brow_ld * LDS_STRIDE + bk_ld];
      *(float4*)(dst)     = p0;
      *(float4*)(dst + 4) = p1;
    }

    __syncthreads();   // needs only dscnt drain; buf reuse distance is 2
                       // iterations, protected by the barrier chain

    // issue next tile's global loads AFTER the barrier: nothing consumes
    // them until the next iteration's ds_store, so they overlap the
    // entire 32-WMMA chain below (uniform branch, EXEC stays all-1s)
    if (ct + 1 < COL_TILES) {
      const float* src = E + (size_t)(cbase + (ct + 1) * 16 + brow_ld) * DD + bk_ld;
      p0 = *(const float4*)(src);
      p1 = *(const float4*)(src + 4);
      __builtin_prefetch(E + (size_t)(cbase + (ct + 2) * 16 + brow_ld) * DD + bk_ld, 0, 1);
    }

    const int lj = lbl[ct * 16 + lcol];   // LDS read: dscnt, not loadcnt

    const float* bp = &bt[buf][lcol * LDS_STRIDE + koff];
    v8f acc = {};
#pragma unroll
    for (int s = 0; s < 32; ++s) {
      v2f bv;
      bv.x = bp[4 * s];
      bv.y = bp[4 * s + 1];
      acc = __builtin_amdgcn_wmma_f32_16x16x4_f32(false, a[s], false, bv,
                                                  (short)0, acc, false, false);
    }

#pragma unroll
    for (int v = 0; v < 8; ++v) {       // C layout: M = v + 8*lh, N = lcol
      float dist = 1.0f - acc[v];
      bool same = (li[v] == lj);
      posr[v] = same ? fmaxf(posr[v], dist) : posr[v];   // branchless
      negr[v] = same ? negr[v] : fminf(negr[v], dist);
    }
  }

  // fold the 16 columns held across lanes of each half-wave
#pragma unroll
  for (int off = 1; off < 16; off <<= 1) {
#pragma unroll
    for (int v = 0; v < 8; ++v) {
      posr[v] = fmaxf(posr[v], __shfl_xor(posr[v], off, 32));
      negr[v] = fminf(negr[v], __shfl_xor(negr[v], off, 32));
    }
  }

  if (lcol == 0) {   // lanes 0 (rows m0..m0+7) and 16 (rows m0+8..m0+15)
    const int rbase = m0 + 8 * lh;
#pragma unroll
    for (int v = 0; v < 8; ++v) {
      atomicMax(&posU[rbase + v], ford(posr[v]));
      atomicMin(&negU[rbase + v], ford(negr[v]));
    }
  }
}

__global__ void th_finalize(const unsigned* __restrict__ posU,
                            const unsigned* __restrict__ negU,
                            float* __restrict__ out) {
  __shared__ float red[8];
  int i = blockIdx.x * 256 + threadIdx.x;  // grid exactly covers NN
  float hp = fmaxf(forddec(posU[i]), 0.0f);
  float hn = forddec(negU[i]);
  float loss = fmaxf(MARGIN_F + hp - hn, 0.0f);
#pragma unroll
  for (int off = 16; off >= 1; off >>= 1)
    loss += __shfl_xor(loss, off, 32);
  int lane = threadIdx.x & 31, wv = threadIdx.x >> 5;
  if (lane == 0) red[wv] = loss;
  __syncthreads();
  if (threadIdx.x == 0) {
    float s = 0.0f;
#pragma unroll
    for (int w = 0; w < 8; ++w) s += red[w];
    atomicAdd(out, s * (1.0f / (float)NN));
  }
}

extern "C" void kernel_launch(void* const* d_in, const int* in_sizes, int n_in,
                              void* d_out, int out_size, void* d_ws, size_t ws_size,
                              hipStream_t stream) {
  const float* E      = (const float*)d_in[0];
  const int*   labels = (const int*)d_in[1];
  float*       out    = (float*)d_out;
  unsigned*    posU   = (unsigned*)d_ws;
  unsigned*    negU   = posU + NN;

  th_init<<<(NN + 255) / 256, 256, 0, stream>>>(posU, negU, out);
  dim3 grid(NN / ROWS_PER_BLOCK, COL_SPLIT);
  th_main<<<grid, 256, 0, stream>>>(E, labels, posU, negU);
  th_finalize<<<NN / 256, 256, 0, stream>>>(posU, negU, out);
}